// FCOSPostProcessor_4913442586709
// MI455X (gfx1250) — compile-verified
//
#include <hip/hip_runtime.h>
#include <math.h>
#include <stdint.h>

// ---------------------------------------------------------------------------
// FCOS post-processor for MI455X (gfx1250).
// Bandwidth-bound: 2 streaming passes over box_cls+iou (336MB each) at
// 23.3 TB/s ~= 29us total. Top-K via per-image 4096-bin histogram on the
// monotone float bit pattern, then exact in-LDS bitonic sort of <=4096
// candidates per image.
// CDNA5 paths: async global->LDS staging (ASYNCcnt) + WMMA triangular-matmul
// suffix scan on the matrix pipe.
// ---------------------------------------------------------------------------

#define NN   32
#define CC   80
#define TT   16384          // power of two: t = j & 16383, c = j >> 14
#define KK   1000
#define BINS 4096
#define CAP  4096           // per-image candidate capacity
#define TILE 4096           // elements staged per block (16KB f32)
#define TILES ((CC * TT) / TILE)   // 320

// ---- feature guards (device pass only) ------------------------------------
#if defined(__HIP_DEVICE_COMPILE__) && defined(__gfx1250__) && \
    __has_builtin(__builtin_amdgcn_wmma_f32_16x16x4_f32)
#define HAVE_WMMA_SCAN 1
#endif

#if defined(__HIP_DEVICE_COMPILE__) && defined(__gfx1250__) && \
    __has_builtin(__builtin_amdgcn_global_load_async_to_lds_b128) && \
    __has_builtin(__builtin_amdgcn_s_wait_asynccnt)
#define HAVE_ASYNC_LDS 1
#endif

typedef float v8f __attribute__((ext_vector_type(8)));
typedef float v2f __attribute__((ext_vector_type(2)));

#if defined(HAVE_ASYNC_LDS)
// The builtin's params are pointers to int __vector(4) in AS(1)/AS(3)
// (clang prints HIP AS1 as `__device__`). Match the pointee type exactly
// with gcc-style vector_size, and reach the address-space-qualified pointer
// types via integer casts (generic shared pointers carry the LDS byte offset
// in their low 32 bits on AMDGPU, so truncation yields a valid AS3 address).
typedef int v4i_vs __attribute__((vector_size(4 * sizeof(int))));
typedef __attribute__((address_space(1))) v4i_vs as1_v4i;
typedef __attribute__((address_space(3))) v4i_vs as3_v4i;

__device__ __forceinline__ void async_b128(const float* g, const float* l) {
  __builtin_amdgcn_global_load_async_to_lds_b128(
      (as1_v4i*)(uintptr_t)g,
      (as3_v4i*)(unsigned)(uintptr_t)l, 0, 0);
}
#endif

__device__ __forceinline__ void stage_tile(const float* g, float* l, int tid) {
#if defined(HAVE_ASYNC_LDS)
#pragma unroll
  for (int u = 0; u < 4; ++u) {
    const int f = u * 1024 + tid * 4;   // 256 lanes * 16B, fully coalesced
    async_b128(g + f, l + f);
  }
#else
#pragma unroll
  for (int u = 0; u < 16; ++u) {
    const int e = u * 256 + tid;
    l[e] = g[e];
  }
#endif
}

__device__ __forceinline__ void stage_wait() {
#if defined(HAVE_ASYNC_LDS)
  __builtin_amdgcn_s_wait_asynccnt(0);
#endif
}

__device__ __forceinline__ float sigmoidf_(float x) {
  return 1.0f / (1.0f + __expf(-x));
}

// ---------------------------------------------------------------------------
// Pass 1: score + per-image histogram of float-ordered bits (4096 bins).
// grid = (TILES, N), block = 256
// ---------------------------------------------------------------------------
__global__ void fcos_score_hist(const float* __restrict__ box_cls,
                                const float* __restrict__ iou,
                                unsigned* __restrict__ ghist) {
  const int n   = blockIdx.y;
  const int tid = threadIdx.x;
  __shared__ unsigned hist[BINS];
  __shared__ float lcls[TILE];
  __shared__ float liou[TILE];

  for (int i = tid; i < BINS; i += 256) hist[i] = 0u;

  const long base = (long)n * CC * TT + (long)blockIdx.x * TILE;
  stage_tile(box_cls + base, lcls, tid);
  stage_tile(iou + base, liou, tid);
  stage_wait();
  __syncthreads();

#pragma unroll
  for (int u = 0; u < 16; ++u) {
    const int e = u * 256 + tid;
    const float cs = sigmoidf_(lcls[e]);
    if (cs > 0.05f) {                              // candidate mask (pre-iou)
      const float s = cs * sigmoidf_(liou[e]);     // score in (0,1)
      unsigned bin = __float_as_uint(s) >> 18;     // monotone key -> 14 bits
      if (bin >= BINS) bin = BINS - 1;
      atomicAdd(&hist[bin], 1u);
    }
  }
  __syncthreads();

  unsigned* gh = ghist + n * BINS;
  for (int i = tid; i < BINS; i += 256)
    if (hist[i]) atomicAdd(&gh[i], hist[i]);
}

// ---------------------------------------------------------------------------
// Threshold find: per-image suffix-scan of 4096 bins, locate bin where the
// top-down cumulative count crosses K. The 256 block-sum suffix scan is done
// as a triangular-ones matmul on the matrix pipe (counts <= 2^24, exact in
// f32): D = U(16x16, upper-tri incl diag) x X(16x16 block sums) gives the
// within-column inclusive suffixes; cross-column offsets are a 16-long scalar
// suffix. grid = N, block = 256
// ---------------------------------------------------------------------------
__global__ void fcos_find_thresh(const unsigned* __restrict__ ghist,
                                 unsigned* __restrict__ selBin,
                                 unsigned* __restrict__ cntAbove) {
  const int n   = blockIdx.x;
  const int tid = threadIdx.x;
  __shared__ float bs[256];     // block sums, X[k][c] = bs[c*16+k]
  __shared__ float sfx[256];    // inclusive suffix per 16-bin block

  const unsigned* h = ghist + n * BINS;
  unsigned localh[16];
  float mysum = 0.f;
#pragma unroll
  for (int w = 0; w < 16; ++w) {
    localh[w] = h[tid * 16 + w];
    mysum += (float)localh[w];
  }
  bs[tid] = mysum;
  __syncthreads();

#if defined(HAVE_WMMA_SCAN)
  __shared__ float colsfx[16];
  if (tid < 32) {                     // wave 0, EXEC all ones
    const int lane  = tid;
    const int m     = lane & 15;      // A row / D column index
    const int khalf = (lane < 16) ? 0 : 2;
    v8f d = {0.f, 0.f, 0.f, 0.f, 0.f, 0.f, 0.f, 0.f};
#pragma unroll
    for (int k0 = 0; k0 < 4; ++k0) {
      const int ka = k0 * 4 + khalf;
      v2f a, b;
      a.x = (ka >= m) ? 1.f : 0.f;        // U[m][ka]
      a.y = (ka + 1 >= m) ? 1.f : 0.f;    // U[m][ka+1]
      b.x = bs[m * 16 + ka];              // X[ka][m]
      b.y = bs[m * 16 + ka + 1];          // X[ka+1][m]
      d = __builtin_amdgcn_wmma_f32_16x16x4_f32(
          false, a, false, b, (short)0, d, false, false);
    }
#pragma unroll
    for (int v = 0; v < 8; ++v) {
      const int row = v + ((lane >= 16) ? 8 : 0);   // D row (M)
      sfx[m * 16 + row] = d[v];                     // block i = c*16 + r
    }
  }
  __syncthreads();
  if (tid == 0) {           // cross-column exclusive suffix from D row 0
    float run = 0.f;
    for (int c2 = 15; c2 >= 0; --c2) {
      colsfx[c2] = run;
      run += sfx[c2 * 16 + 0];      // column total = (UxX)[0][c]
    }
  }
  __syncthreads();
  const float incl = sfx[tid] + colsfx[tid >> 4];
#else
  if (tid == 0) {
    float run = 0.f;
    for (int i = 255; i >= 0; --i) { run += bs[i]; sfx[i] = run; }
  }
  __syncthreads();
  const float incl = sfx[tid];
#endif

  // strictly-above count for bins higher than this thread's block
  unsigned above = (unsigned)(incl - mysum);
  for (int w = 15; w >= 0; --w) {       // walk own 16 bins top-down
    const unsigned hb = localh[w];
    if (above < (unsigned)KK && above + hb >= (unsigned)KK) {
      selBin[n]   = (unsigned)(tid * 16 + w);   // unique writer
      cntAbove[n] = above;
    }
    above += hb;
  }
}

// ---------------------------------------------------------------------------
// Pass 2: recompute scores, collect candidates in bins >= threshold bin.
// grid = (TILES, N), block = 256
// ---------------------------------------------------------------------------
__global__ void fcos_collect(const float* __restrict__ box_cls,
                             const float* __restrict__ iou,
                             const unsigned* __restrict__ selBin,
                             unsigned* __restrict__ cnt,
                             unsigned long long* __restrict__ cand) {
  const int n   = blockIdx.y;
  const int tid = threadIdx.x;
  __shared__ float lcls[TILE];
  __shared__ float liou[TILE];

  const long base = (long)n * CC * TT + (long)blockIdx.x * TILE;
  stage_tile(box_cls + base, lcls, tid);
  stage_tile(iou + base, liou, tid);
  stage_wait();
  __syncthreads();

  const unsigned B = selBin[n];
#pragma unroll
  for (int u = 0; u < 16; ++u) {
    const int e = u * 256 + tid;
    const float cs = sigmoidf_(lcls[e]);
    if (cs > 0.05f) {
      const float s = cs * sigmoidf_(liou[e]);
      const unsigned key = __float_as_uint(s);
      if ((key >> 18) >= B) {
        const unsigned pos = atomicAdd(&cnt[n], 1u);
        if (pos < CAP) {
          const int j = blockIdx.x * TILE + e;      // layout [C,T]
          const int t = j & (TT - 1);
          const int c = j >> 14;
          const unsigned refIdx = (unsigned)(t * CC + c);  // [T,C] flat idx
          // sort key: score desc, then refIdx asc (matches top_k tie-break)
          cand[(long)n * CAP + pos] =
              ((unsigned long long)key << 32) | (unsigned)(~refIdx);
        }
      }
    }
  }
}

// ---------------------------------------------------------------------------
// Per-image bitonic sort (4096 entries in 32KB LDS) + decode/emit.
// grid = N, block = 1024
// ---------------------------------------------------------------------------
__global__ void fcos_sort_emit(const unsigned long long* __restrict__ cand,
                               const unsigned* __restrict__ cnt,
                               const float* __restrict__ locations,
                               const float* __restrict__ box_reg,
                               float* __restrict__ out) {
  const int n   = blockIdx.x;
  const int tid = threadIdx.x;
  __shared__ unsigned long long buf[CAP];

  unsigned stored = cnt[n];
  if (stored > CAP) stored = CAP;
  for (int i = tid; i < CAP; i += 1024)
    buf[i] = (i < (int)stored) ? cand[(long)n * CAP + i] : 0ull;
  __syncthreads();

  // bitonic sort, descending (zeros sink to the end)
  for (unsigned k = 2; k <= CAP; k <<= 1) {
    for (unsigned j = k >> 1; j > 0; j >>= 1) {
      for (unsigned i = tid; i < CAP; i += 1024) {
        const unsigned ixj = i ^ j;
        if (ixj > i) {
          const unsigned long long a = buf[i];
          const unsigned long long b = buf[ixj];
          const bool desc = ((i & k) == 0);
          if (desc ? (a < b) : (a > b)) { buf[i] = b; buf[ixj] = a; }
        }
      }
      __syncthreads();
    }
  }

  if (tid < KK) {
    const unsigned long long p = buf[tid];
    const bool valid = (tid < (int)stored) && (p != 0ull);
    const unsigned key    = (unsigned)(p >> 32);
    const unsigned refIdx = ~(unsigned)(p & 0xffffffffu);
    const int t  = (int)(refIdx / CC);
    const int cl = (int)(refIdx % CC);

    const float score = __uint_as_float(key);
    const float loc   = locations[t];
    const float reg0  = box_reg[(long)n * 2 * TT + t];
    const float reg1  = box_reg[(long)n * 2 * TT + TT + t];
    const float start = fminf(fmaxf((loc - reg0) * (1.0f / 32.0f), 0.f), 1.f);
    const float end   = fminf(fmaxf((loc + reg1) * (1.0f / 32.0f), 0.f), 1.f);
    const bool keep   = valid && ((end - start) >= 0.0f);

    // output layout (flat f32): det[N,K,2] | scores[N,K] | labels[N,K] |
    //                           locs[N,K] | keep[N,K]
    const int NK = NN * KK;
    const int o  = n * KK + tid;
    out[2 * o + 0]          = keep ? start : 0.f;
    out[2 * o + 1]          = keep ? end : 0.f;
    out[2 * NK + o]         = valid ? sqrtf(score) : 0.f;
    out[3 * NK + o]         = valid ? (float)(cl + 1) : 0.f;
    out[4 * NK + o]         = valid ? loc * (1.0f / 32.0f) : 0.f;
    out[5 * NK + o]         = keep ? 1.f : 0.f;
  }
}

// ---------------------------------------------------------------------------
// Host launcher. Workspace layout (bytes):
//   [0,      524288)  per-image histograms  32*4096 u32
//   [524288, 524416)  cnt[32]
//   [524416, 524544)  selBin[32]
//   [524544, 524672)  cntAbove[32]
//   [524800, 1573376) cand  32*4096 u64
// ---------------------------------------------------------------------------
extern "C" void kernel_launch(void* const* d_in, const int* in_sizes, int n_in,
                              void* d_out, int out_size, void* d_ws,
                              size_t ws_size, hipStream_t stream) {
  (void)in_sizes; (void)n_in; (void)out_size; (void)ws_size;
  const float* locations = (const float*)d_in[0];
  const float* box_cls   = (const float*)d_in[1];
  const float* box_reg   = (const float*)d_in[2];
  const float* iou       = (const float*)d_in[3];
  float* out = (float*)d_out;

  char* ws = (char*)d_ws;
  unsigned* ghist           = (unsigned*)(ws);
  unsigned* cnt             = (unsigned*)(ws + 524288);
  unsigned* selB            = (unsigned*)(ws + 524416);
  unsigned* cAbv            = (unsigned*)(ws + 524544);
  unsigned long long* cand  = (unsigned long long*)(ws + 524800);

  // zero histograms + counters every call (graph-capture safe)
  (void)hipMemsetAsync(ws, 0, 524672, stream);

  dim3 gridS(TILES, NN);
  fcos_score_hist<<<gridS, 256, 0, stream>>>(box_cls, iou, ghist);
  fcos_find_thresh<<<NN, 256, 0, stream>>>(ghist, selB, cAbv);
  fcos_collect<<<gridS, 256, 0, stream>>>(box_cls, iou, selB, cnt, cand);
  fcos_sort_emit<<<NN, 1024, 0, stream>>>(cand, cnt, locations, box_reg, out);
}